// CopyDecoder_59442347376667
// MI455X (gfx1250) — compile-verified
//
#include <hip/hip_runtime.h>
#include <cstdint>

// ---------------------------------------------------------------------------
// CopyDecoder step for MI455X (gfx1250, wave32).
// Memory-bound workload (~350 MiB traffic, ~37 GFLOP) -> fp32 WMMA
// (V_WMMA_F32_16X16X4_F32) for all GEMMs; async-to-LDS staging for the
// encoder_states tiles in the fused attention-score kernel.
// ---------------------------------------------------------------------------

#define V_  32000
#define E_  256
#define H_  512
#define B_  64
#define S_  512
#define TWO_H  1024
#define THREE_H 1536
#define EK  (E_ + TWO_H)   // 1280

typedef float v2f __attribute__((ext_vector_type(2)));
typedef float v8f __attribute__((ext_vector_type(8)));
typedef int   v4i __attribute__((ext_vector_type(4)));

#define AS1 __attribute__((address_space(1)))
#define AS3 __attribute__((address_space(3)))

#ifndef __has_builtin
#define __has_builtin(x) 0
#endif
#if __has_builtin(__builtin_amdgcn_global_load_async_to_lds_b128)
#define USE_ASYNC_LDS 1
#else
#define USE_ASYNC_LDS 0
#endif

#if USE_ASYNC_LDS
__device__ __forceinline__ void async_copy_b128(const float* g, float* l) {
  // per-lane 16B global -> LDS transfer, tracked by ASYNCcnt
  __builtin_amdgcn_global_load_async_to_lds_b128(
      (AS1 v4i*)(v4i*)(void*)g, (AS3 v4i*)(v4i*)(void*)l, 0, 0);
}
__device__ __forceinline__ void wait_async0() {
#if __has_builtin(__builtin_amdgcn_s_wait_asynccnt)
  __builtin_amdgcn_s_wait_asynccnt(0);
#else
  asm volatile("s_wait_asynccnt 0" ::: "memory");
#endif
}
#endif

// ---------------------------------------------------------------------------
// K0: x = concat(embed[inputs], weighted)   (B x 1280)
// ---------------------------------------------------------------------------
__global__ void build_x_kernel(const int* __restrict__ inputs,
                               const float* __restrict__ embed,
                               const float* __restrict__ weighted,
                               float* __restrict__ x) {
  int idx = blockIdx.x * blockDim.x + threadIdx.x;   // B*EK = 81920
  if (idx >= B_ * EK) return;
  int b = idx / EK, k = idx - b * EK;
  x[idx] = (k < E_) ? embed[(size_t)inputs[b] * E_ + k]
                    : weighted[b * TWO_H + (k - E_)];
}

// ---------------------------------------------------------------------------
// Generic fp32 WMMA GEMM:  C[M x N] = A[M x K] @ B[K x N] + bias[N]
// One wave per 16x16 output tile. grid = (N/16, M/16), block = 32.
// A-frag layout (16x4 f32): lane&15 = M, lane>>4 selects K pair {0,1}/{2,3}.
// B-frag layout (4x16 f32): lane&15 = N, lane>>4 selects K pair.
// D layout (16x16 f32): VGPR i -> row i + 8*(lane>>4), col = lane&15.
// ---------------------------------------------------------------------------
__global__ __launch_bounds__(32)
void wmma_gemm_bias_kernel(const float* __restrict__ A, int lda,
                           const float* __restrict__ Bm, int ldb,
                           const float* __restrict__ bias,
                           float* __restrict__ C, int ldc, int K) {
  const int nt = blockIdx.x, mt = blockIdx.y;
  const int lane = threadIdx.x;
  const int half = lane >> 4, q = lane & 15;
  const int m = mt * 16 + q;
  const int n = nt * 16 + q;

  v8f acc = {};
  const v2f* arow2 = (const v2f*)(A + (size_t)m * lda);
  for (int k0 = 0; k0 < K; k0 += 4) {
    const int ka = k0 + 2 * half;           // even
    v2f a = arow2[ka >> 1];                 // b64 load of A[m][ka..ka+1]
    v2f b;
    b[0] = Bm[(size_t)ka * ldb + n];
    b[1] = Bm[(size_t)(ka + 1) * ldb + n];
    acc = __builtin_amdgcn_wmma_f32_16x16x4_f32(false, a, false, b,
                                                (short)0, acc, false, false);
  }
  const float bv = bias ? bias[n] : 0.0f;
  #pragma unroll
  for (int i = 0; i < 8; ++i) {
    int row = mt * 16 + i + 8 * half;
    C[(size_t)row * ldc + n] = acc[i] + bv;
  }
}

// ---------------------------------------------------------------------------
// K3: GRU gates -> state  (writes the `state` output)
// ---------------------------------------------------------------------------
__global__ void gru_gates_kernel(const float* __restrict__ mx,
                                 const float* __restrict__ mh,
                                 const float* __restrict__ prev,
                                 float* __restrict__ state) {
  int idx = blockIdx.x * blockDim.x + threadIdx.x;   // B*H
  if (idx >= B_ * H_) return;
  int b = idx >> 9, h = idx & (H_ - 1);
  const float* mxr = mx + (size_t)b * THREE_H;
  const float* mhr = mh + (size_t)b * THREE_H;
  float z  = 1.0f / (1.0f + __expf(-(mxr[h]           + mhr[h])));
  float r  = 1.0f / (1.0f + __expf(-(mxr[H_ + h]      + mhr[H_ + h])));
  float hh = tanhf(mxr[2 * H_ + h] + r * mhr[2 * H_ + h]);
  state[idx] = z * prev[idx] + (1.0f - z) * hh;
}

// ---------------------------------------------------------------------------
// K5: fused attention score:
//   sc = tanh(ES[b,s,:] @ Wc_k + Wc_b);  score_c[b,s] = tanh(sc . state[b]) + mask
// One block = 4 waves handles 16 (b,s) rows. ES tile staged in LDS (async).
// Each wave owns 8 N-tiles of H (=128 cols); K=1024 in two 512-chunks.
// ---------------------------------------------------------------------------
__global__ __launch_bounds__(128)
void score_c_kernel(const float* __restrict__ es,
                    const float* __restrict__ Wc_k,
                    const float* __restrict__ Wc_b,
                    const float* __restrict__ state,
                    const int* __restrict__ enc_in,
                    float* __restrict__ score_c) {
  __shared__ float Atile[16 * 512];        // 32 KB: 16 rows x 512 K-slice
  __shared__ float partial[4][16];

  const int tile = blockIdx.x;             // B*S/16 = 2048
  const int b    = tile >> 5;              // 32 s-tiles per batch row
  const int s0   = (tile & 31) << 4;
  const int tid  = threadIdx.x;
  const int lane = tid & 31;
  const int wave = tid >> 5;
  const int half = lane >> 4, q = lane & 15;

  const float* esb = es + ((size_t)b * S_ + s0) * TWO_H;

  v8f acc[8];
  #pragma unroll
  for (int j = 0; j < 8; ++j) acc[j] = (v8f){};

  for (int kc = 0; kc < 2; ++kc) {
    // ---- stage 16 x 512 fp32 A-tile into LDS ----
#if USE_ASYNC_LDS
    #pragma unroll
    for (int i = 0; i < 16; ++i) {
      int chunk = i * 128 + tid;                  // 2048 b128 chunks
      int f = chunk * 4;                          // float offset in tile
      int row = f >> 9, col = f & 511;
      async_copy_b128(esb + (size_t)row * TWO_H + kc * 512 + col, &Atile[f]);
    }
    wait_async0();
#else
    for (int chunk = tid; chunk < 2048; chunk += 128) {
      int f = chunk * 4;
      int row = f >> 9, col = f & 511;
      *(float4*)&Atile[f] = *(const float4*)(esb + (size_t)row * TWO_H + kc * 512 + col);
    }
#endif
    __syncthreads();

    // ---- WMMA over this K chunk for this wave's 8 N-tiles ----
    const v2f* A2 = (const v2f*)Atile;
    #pragma unroll
    for (int j = 0; j < 8; ++j) {
      const int nt = wave * 8 + j;
      const int n  = nt * 16 + q;
      for (int k0 = 0; k0 < 512; k0 += 4) {
        const int ka = k0 + 2 * half;
        v2f a = A2[(q * 512 + ka) >> 1];
        v2f bb;
        const size_t krow = (size_t)(kc * 512 + ka) * H_;
        bb[0] = Wc_k[krow + n];
        bb[1] = Wc_k[krow + H_ + n];
        acc[j] = __builtin_amdgcn_wmma_f32_16x16x4_f32(false, a, false, bb,
                                                       (short)0, acc[j],
                                                       false, false);
      }
    }
    __syncthreads();
  }

  // ---- epilogue: tanh(sc)*state, reduce over cols ----
  float rowsum[8];
  #pragma unroll
  for (int i = 0; i < 8; ++i) rowsum[i] = 0.0f;
  #pragma unroll
  for (int j = 0; j < 8; ++j) {
    const int n = (wave * 8 + j) * 16 + q;
    const float sb = Wc_b[n];
    const float st = state[(size_t)b * H_ + n];
    #pragma unroll
    for (int i = 0; i < 8; ++i)
      rowsum[i] += tanhf(acc[j][i] + sb) * st;
  }
  // butterfly reduce across the 16 lanes of each half (cols)
  #pragma unroll
  for (int off = 1; off < 16; off <<= 1)
    #pragma unroll
    for (int i = 0; i < 8; ++i)
      rowsum[i] += __shfl_xor(rowsum[i], off, 16);

  if (q == 0) {
    #pragma unroll
    for (int i = 0; i < 8; ++i)
      partial[wave][half * 8 + i] = rowsum[i];
  }
  __syncthreads();
  if (tid < 16) {
    float v = partial[0][tid] + partial[1][tid] + partial[2][tid] + partial[3][tid];
    int s = s0 + tid;
    float msk = (enc_in[(size_t)b * S_ + s] == 0) ? -1000.0f : 0.0f;
    score_c[(size_t)b * S_ + s] = tanhf(v) + msk;
  }
}

// ---------------------------------------------------------------------------
// K6: softmax over concat(score_g[V], score_c[S]) per batch row.
// Writes prob_g into d_out, prob_c in place over score_c.
// ---------------------------------------------------------------------------
__global__ __launch_bounds__(256)
void softmax_kernel(const float* __restrict__ score_g,
                    float* __restrict__ score_c,
                    float* __restrict__ out) {
  const int b = blockIdx.x, tid = threadIdx.x;
  __shared__ float red[256];
  const float* sg = score_g + (size_t)b * V_;
  float* scr = score_c + (size_t)b * S_;

  float m = -3.4e38f;
  for (int i = tid; i < V_; i += 256) m = fmaxf(m, sg[i]);
  for (int i = tid; i < S_; i += 256) m = fmaxf(m, scr[i]);
  red[tid] = m; __syncthreads();
  for (int o = 128; o > 0; o >>= 1) {
    if (tid < o) red[tid] = fmaxf(red[tid], red[tid + o]);
    __syncthreads();
  }
  m = red[0]; __syncthreads();

  float s = 0.0f;
  for (int i = tid; i < V_; i += 256) s += __expf(sg[i] - m);
  for (int i = tid; i < S_; i += 256) s += __expf(scr[i] - m);
  red[tid] = s; __syncthreads();
  for (int o = 128; o > 0; o >>= 1) {
    if (tid < o) red[tid] += red[tid + o];
    __syncthreads();
  }
  const float inv = 1.0f / red[0];

  float* og = out + (size_t)b * V_;
  for (int i = tid; i < V_; i += 256) og[i] = __expf(sg[i] - m) * inv;
  for (int i = tid; i < S_; i += 256) scr[i] = __expf(scr[i] - m) * inv;
}

// ---------------------------------------------------------------------------
// K7: hit flag, attn = prob_c * hit, deterministic serial scatter-add
//     out[b, enc_in[b,s]] += prob_c[b,s]
// ---------------------------------------------------------------------------
__global__ __launch_bounds__(256)
void scatter_kernel(const int* __restrict__ enc_in,
                    const int* __restrict__ inputs,
                    const float* __restrict__ prob_c,
                    float* __restrict__ out,
                    float* __restrict__ attn) {
  const int b = blockIdx.x, tid = threadIdx.x;
  __shared__ int hit_s;
  if (tid == 0) hit_s = 0;
  __syncthreads();

  const int tok = inputs[b];
  const int* ei = enc_in + (size_t)b * S_;
  int local = 0;
  for (int s = tid; s < S_; s += 256) local |= (ei[s] == tok);
  if (local) atomicOr(&hit_s, 1);
  __syncthreads();
  const float hit = hit_s ? 1.0f : 0.0f;

  const float* pc = prob_c + (size_t)b * S_;
  for (int s = tid; s < S_; s += 256) attn[(size_t)b * S_ + s] = pc[s] * hit;

  if (tid == 0) {               // serial per row: deterministic accumulation
    float* og = out + (size_t)b * V_;
    for (int s = 0; s < S_; ++s) og[ei[s]] += pc[s];
  }
}

// ---------------------------------------------------------------------------
// K8: weighted_new[b,k] = sum_s attn[b,s] * encoder_states[b,s,k]
// ---------------------------------------------------------------------------
__global__ __launch_bounds__(256)
void weighted_new_kernel(const float* __restrict__ attn,
                         const float* __restrict__ es,
                         float* __restrict__ wn) {
  const int b = blockIdx.x, tid = threadIdx.x;
  __shared__ float a_s[S_];
  const float* a = attn + (size_t)b * S_;
  for (int i = tid; i < S_; i += 256) a_s[i] = a[i];
  __syncthreads();

  const float4* e = (const float4*)(es + (size_t)b * S_ * TWO_H);
  float4 acc = {0.f, 0.f, 0.f, 0.f};
  for (int s = 0; s < S_; ++s) {
    __builtin_prefetch(&e[(size_t)(s + 8) * 256 + tid], 0, 1);
    const float w = a_s[s];
    float4 v = e[(size_t)s * 256 + tid];
    acc.x += w * v.x; acc.y += w * v.y; acc.z += w * v.z; acc.w += w * v.w;
  }
  ((float4*)(wn + (size_t)b * TWO_H))[tid] = acc;
}

// ---------------------------------------------------------------------------
// Host orchestration
// ---------------------------------------------------------------------------
extern "C" void kernel_launch(void* const* d_in, const int* in_sizes, int n_in,
                              void* d_out, int out_size, void* d_ws, size_t ws_size,
                              hipStream_t stream) {
  const int*   inputs      = (const int*)  d_in[0];
  const float* enc_states  = (const float*)d_in[1];
  const int*   enc_inputs  = (const int*)  d_in[2];
  const float* prev_state  = (const float*)d_in[3];
  const float* weighted    = (const float*)d_in[4];
  const float* embed       = (const float*)d_in[5];
  const float* gru_kernel  = (const float*)d_in[6];
  const float* gru_rkernel = (const float*)d_in[7];
  const float* gru_bias    = (const float*)d_in[8];
  const float* Wo_k        = (const float*)d_in[9];
  const float* Wo_b        = (const float*)d_in[10];
  const float* Wc_k        = (const float*)d_in[11];
  const float* Wc_b        = (const float*)d_in[12];

  // outputs, concatenated flat in return order
  float* out_probs = (float*)d_out;                       // B*V
  float* out_state = out_probs + (size_t)B_ * V_;         // B*H
  float* out_wnew  = out_state + (size_t)B_ * H_;         // B*2H

  // workspace (~9.6 MB of fp32)
  float* ws      = (float*)d_ws;
  float* x_buf   = ws;                                    // B*1280
  float* mx      = x_buf + (size_t)B_ * EK;               // B*1536
  float* mh      = mx + (size_t)B_ * THREE_H;             // B*1536
  float* score_g = mh + (size_t)B_ * THREE_H;             // B*V
  float* score_c = score_g + (size_t)B_ * V_;             // B*S (then prob_c)
  float* attn    = score_c + (size_t)B_ * S_;             // B*S

  // 0) gather x
  build_x_kernel<<<(B_ * EK + 255) / 256, 256, 0, stream>>>(inputs, embed, weighted, x_buf);
  // 1) mx = x @ gru_kernel + gru_bias[0]
  wmma_gemm_bias_kernel<<<dim3(THREE_H / 16, B_ / 16), 32, 0, stream>>>(
      x_buf, EK, gru_kernel, THREE_H, gru_bias, mx, THREE_H, EK);
  // 2) mh = prev_state @ gru_rkernel + gru_bias[1]
  wmma_gemm_bias_kernel<<<dim3(THREE_H / 16, B_ / 16), 32, 0, stream>>>(
      prev_state, H_, gru_rkernel, THREE_H, gru_bias + THREE_H, mh, THREE_H, H_);
  // 3) GRU gates -> state output
  gru_gates_kernel<<<(B_ * H_ + 255) / 256, 256, 0, stream>>>(mx, mh, prev_state, out_state);
  // 4) score_g = state @ Wo_k + Wo_b
  wmma_gemm_bias_kernel<<<dim3(V_ / 16, B_ / 16), 32, 0, stream>>>(
      out_state, H_, Wo_k, V_, Wo_b, score_g, V_, H_);
  // 5) fused attention score (WMMA + async LDS staging), includes -1000 mask
  score_c_kernel<<<(B_ * S_) / 16, 128, 0, stream>>>(
      enc_states, Wc_k, Wc_b, out_state, enc_inputs, score_c);
  // 6) joint softmax -> prob_g (out) and prob_c (in place)
  softmax_kernel<<<B_, 256, 0, stream>>>(score_g, score_c, out_probs);
  // 7) hit/attn + deterministic scatter-add of prob_c into out
  scatter_kernel<<<B_, 256, 0, stream>>>(enc_inputs, inputs, score_c, out_probs, attn);
  // 8) weighted_new = attn @ encoder_states
  weighted_new_kernel<<<B_, 256, 0, stream>>>(attn, enc_states, out_wnew);

  (void)in_sizes; (void)n_in; (void)out_size; (void)ws_size;
}